// FastHST_22943715295841
// MI455X (gfx1250) — compile-verified
//
#include <hip/hip_runtime.h>
#include <math.h>

typedef __attribute__((ext_vector_type(2))) float v2f;
typedef __attribute__((ext_vector_type(8))) float v8f;
typedef __attribute__((address_space(3))) float lds_f;

// D = A(16x4) * B(4x16) + C, fp32 WMMA. 8-arg form:
// (neg_a, A, neg_b, B, c_mod, C, reuse_a, reuse_b)
#define WMMAF32(a, b, c) \
  __builtin_amdgcn_wmma_f32_16x16x4_f32(false, (a), false, (b), (short)0, (c), false, false)

static __device__ __forceinline__ v8f vzero8() {
  v8f z = {0.f, 0.f, 0.f, 0.f, 0.f, 0.f, 0.f, 0.f};
  return z;
}

// CDNA5 async global->LDS copy, 16B per lane, ASYNCcnt-tracked (ISA 15.18.3 op 98).
static __device__ __forceinline__ void async_ld_b128(const float* g, float* l) {
  unsigned ldsaddr = (unsigned)(unsigned long long)(lds_f*)l;  // generic -> AS3 -> 32-bit LDS addr
  asm volatile("global_load_async_to_lds_b128 %0, %1, off"
               :
               : "v"(ldsaddr), "v"((unsigned long long)g)
               : "memory");
}

static __device__ __forceinline__ void wait_async0() {
#if __has_builtin(__builtin_amdgcn_s_wait_asynccnt)
  __builtin_amdgcn_s_wait_asynccnt(0);
#else
  asm volatile("s_wait_asynccnt 0x0" ::: "memory");
#endif
}

// Per-lane B operands for the two Toeplitz matrices of a 16-tap filter.
// Assumed B layout (mirror of documented A layout): lane holds column n=L&15,
// K rows split as kb=(L>>4)*2 across lane halves, consecutive K in the v2f pair.
static __device__ __forceinline__ void make_b(const float* __restrict__ w, int n, int kb,
                                              v2f* __restrict__ b0, v2f* __restrict__ b1) {
#pragma unroll
  for (int c = 0; c < 4; ++c) {
    int d0 = 4 * c + kb;
    int d1 = d0 + 1;
    b0[c].x = (d0 >= n) ? w[d0 - n] : 0.0f;   // W0[d][n] = w[d-n], d>=n
    b0[c].y = (d1 >= n) ? w[d1 - n] : 0.0f;
    int e0 = 16 + d0 - n;                     // W1[d][n] = w[16+d-n], 16+d-n<16
    b1[c].x = (e0 < 16) ? w[e0] : 0.0f;
    b1[c].y = (e0 + 1 < 16) ? w[e0 + 1] : 0.0f;
  }
}

// One real-input level (levels 1..3): 3 filter chains per 256-output window.
static __device__ __forceinline__ void level_real(
    const float* __restrict__ xin, float* __restrict__ xnext, float* __restrict__ out,
    int wave, int n, int kb, int mhalf, bool evenlane,
    const v2f* bph0, const v2f* bph1, const v2f* bpr0, const v2f* bpr1,
    const v2f* bpi0, const v2f* bpi1,
    int nwin, int p, int a, int core, int Lnext, int nextstart, int nextalloc,
    int outbase, int srow) {
  for (int win = wave; win < nwin; win += 4) {
    const int ib = win << 8;
    v8f cf = vzero8(), cr = vzero8(), ci = vzero8();
#pragma unroll
    for (int cc = 0; cc < 4; ++cc) {
      const int i0 = ib + (n << 4) + (cc << 2) + kb;
      v2f a0 = {xin[i0], xin[i0 + 1]};
      v2f a1 = {xin[i0 + 16], xin[i0 + 17]};
      cf = WMMAF32(a0, bph0[cc], cf);
      cf = WMMAF32(a1, bph1[cc], cf);
      cr = WMMAF32(a0, bpr0[cc], cr);
      cr = WMMAF32(a1, bpr1[cc], cr);
      ci = WMMAF32(a0, bpi0[cc], ci);
      ci = WMMAF32(a1, bpi1[cc], ci);
    }
    if (evenlane) {  // only even outputs survive the stride-2 decimation
#pragma unroll
      for (int v = 0; v < 8; ++v) {
        const int t = p + ib + ((v + mhalf) << 4) + n;
        if (t >= a && t < a + core) {
          v2f s = {cf[v], 0.0f};  // imag(s)=0 for real-input levels
          *(v2f*)(out + outbase + 2 * (srow + (t >> 1))) = s;
        }
        if (xnext) {
          const float wmag = sqrtf(cr[v] * cr[v] + ci[v] * ci[v]);
          const int i2 = t >> 1;
          const int off = i2 - nextstart;
          if (off >= 0 && off < nextalloc && i2 >= 0 && i2 < Lnext) xnext[off] = wmag;
        }
      }
    }
  }
}

__global__ __launch_bounds__(128) void hst_fused_kernel(
    const float* __restrict__ xr, const float* __restrict__ xi,
    const float* __restrict__ psr, const float* __restrict__ psi,
    const float* __restrict__ phi, float* __restrict__ out) {
  // LDS: x0r[4608] x0i[4608] x1[2320] x2[1296] x3[528] = 53.4 KB
  __shared__ float lds[13360];
  float* x0r = lds;
  float* x0i = lds + 4608;
  float* x1 = lds + 9216;
  float* x2 = x1 + 2320;
  float* x3 = x2 + 1296;

  const int T = 524288;
  const int row = blockIdx.y;
  const int c0 = (int)blockIdx.x * 4096;
  const float* xrrow = xr + (size_t)row * T;
  const float* xirow = xi + (size_t)row * T;

  // --- Stage level-0 chunk + halo into LDS.
  // Interior chunks: uniform async global->LDS B128 transfers (9 x 128 lanes x 16B
  // covers [c0-120, c0+4488), inside neighbor chunks). Edge chunks: guarded loads.
  const bool interior = (blockIdx.x > 0) && (blockIdx.x < 127);
  if (interior) {
    const float* gr = xrrow + (c0 - 120);
    const float* gi = xirow + (c0 - 120);
#pragma unroll
    for (int it = 0; it < 9; ++it) {
      const int fo = it * 512 + (int)threadIdx.x * 4;
      async_ld_b128(gr + fo, x0r + fo);
      async_ld_b128(gi + fo, x0i + fo);
    }
  } else {
    for (int idx = threadIdx.x; idx < 4368; idx += 128) {
      const int g = c0 - 120 + idx;
      const bool ok = (g >= 0) && (g < T);
      x0r[idx] = ok ? xrrow[g] : 0.0f;
      x0i[idx] = ok ? xirow[g] : 0.0f;
    }
  }
  // Zero-init deeper-level buffers (edge padding + unwritten tails).
  for (int idx = threadIdx.x; idx < (2320 + 1296 + 528); idx += 128) x1[idx] = 0.0f;

  const int lane = threadIdx.x & 31;
  const int wave = threadIdx.x >> 5;
  const int n = lane & 15;
  const int kb = (lane >> 4) << 1;     // K half select for A/B operands
  const int mhalf = (lane >> 4) << 3;  // C/D row half (ISA-documented layout)
  const bool evenlane = ((n & 1) == 0);

  v2f bph0[4], bph1[4], bpr0[4], bpr1[4], bpi0[4], bpi1[4];
  make_b(phi, n, kb, bph0, bph1);
  make_b(psr, n, kb, bpr0, bpr1);
  make_b(psi, n, kb, bpi0, bpi1);

  wait_async0();
  __syncthreads();

  // ---- Level 0: complex input, 6 conv chains, 17 windows over [c0-112, c0+4240)
  {
    const int p0 = c0 - 112;
    const int x1start = (c0 >> 1) - 56;  // abs level-1 index of x1[0]
    for (int win = wave; win < 17; win += 4) {
      const int ib = win << 8;
      v8f cfr = vzero8(), cfi = vzero8();
      v8f crr = vzero8(), cri = vzero8(), cir = vzero8(), cii = vzero8();
#pragma unroll
      for (int cc = 0; cc < 4; ++cc) {
        const int i0 = ib + (n << 4) + (cc << 2) + kb;
        v2f a0r = {x0r[i0], x0r[i0 + 1]};
        v2f a1r = {x0r[i0 + 16], x0r[i0 + 17]};
        v2f a0i = {x0i[i0], x0i[i0 + 1]};
        v2f a1i = {x0i[i0 + 16], x0i[i0 + 17]};
        cfr = WMMAF32(a0r, bph0[cc], cfr);
        cfr = WMMAF32(a1r, bph1[cc], cfr);
        cfi = WMMAF32(a0i, bph0[cc], cfi);
        cfi = WMMAF32(a1i, bph1[cc], cfi);
        crr = WMMAF32(a0r, bpr0[cc], crr);
        crr = WMMAF32(a1r, bpr1[cc], crr);
        cri = WMMAF32(a0r, bpi0[cc], cri);
        cri = WMMAF32(a1r, bpi1[cc], cri);
        cir = WMMAF32(a0i, bpr0[cc], cir);
        cir = WMMAF32(a1i, bpr1[cc], cir);
        cii = WMMAF32(a0i, bpi0[cc], cii);
        cii = WMMAF32(a1i, bpi1[cc], cii);
      }
      if (evenlane) {
#pragma unroll
        for (int v = 0; v < 8; ++v) {
          const int t = p0 + ib + ((v + mhalf) << 4) + n;
          if (t >= c0 && t < c0 + 4096) {  // s0 (complex) coefficient
            v2f s = {cfr[v], cfi[v]};
            *(v2f*)(out + 2 * (row * 262144 + (t >> 1))) = s;
          }
          const float ure = crr[v] - cii[v];
          const float uim = cri[v] + cir[v];
          const float wmag = sqrtf(ure * ure + uim * uim);
          const int i1 = t >> 1;
          const int off = i1 - x1start;
          if (off >= 0 && off < 2320 && i1 >= 0 && i1 < 262144) x1[off] = wmag;
        }
      }
    }
  }
  __syncthreads();

  const int a1 = c0 >> 1, a2 = c0 >> 2, a3 = c0 >> 3;

  // ---- Level 1: 9 windows; s1 -> out section, |u1| -> x2
  level_real(x1, x2, out, wave, n, kb, mhalf, evenlane,
             bph0, bph1, bpr0, bpr1, bpi0, bpi1,
             9, a1 - 48, a1, 2048, 131072, a2 - 24, 1296,
             16777216, row * 131072);
  __syncthreads();

  // ---- Level 2: 5 windows; s2 -> out, |u2| -> x3
  level_real(x2, x3, out, wave, n, kb, mhalf, evenlane,
             bph0, bph1, bpr0, bpr1, bpi0, bpi1,
             5, a2 - 16, a2, 1024, 65536, a3 - 8, 528,
             25165824, row * 65536);
  __syncthreads();

  // ---- Level 3: 2 windows; s3 -> out (no next level)
  level_real(x3, nullptr, out, wave, n, kb, mhalf, evenlane,
             bph0, bph1, bpr0, bpr1, bpi0, bpi1,
             2, a3, a3, 512, 0, 0, 0,
             29360128, row * 32768);
}

extern "C" void kernel_launch(void* const* d_in, const int* in_sizes, int n_in,
                              void* d_out, int out_size, void* d_ws, size_t ws_size,
                              hipStream_t stream) {
  const float* x_real = (const float*)d_in[0];
  const float* x_imag = (const float*)d_in[1];
  const float* psi_real = (const float*)d_in[2];
  const float* psi_imag = (const float*)d_in[3];
  const float* phi = (const float*)d_in[4];
  float* out = (float*)d_out;

  dim3 grid(128, 32);  // 128 chunks of 4096 samples x 32 batch rows
  dim3 block(128);     // 4 waves (wave32)
  hst_fused_kernel<<<grid, block, 0, stream>>>(x_real, x_imag, psi_real, psi_imag, phi, out);
}